// ESMVAEToken_1649267442261
// MI455X (gfx1250) — compile-verified
//
#include <hip/hip_runtime.h>
#include <hip/hip_bf16.h>

typedef float v2f __attribute__((ext_vector_type(2)));
typedef float v8f __attribute__((ext_vector_type(8)));

#define BB 64
#define SS 512
#define DD 1280
#define H1 1024
#define H2 512
#define LL 128
#define VV 33
#define TT 512
#define RH 512
#define G3 1536   // 3*RH
#define GRU_BLOCKS 32
#define WSTRIDE 516   // padded LDS row stride (banks: l*516 % 64 = 4l -> conflict-free)

// fp32 WMMA: D(16x16) = A(16x4) * B(4x16) + C
__device__ __forceinline__ v8f wmma_f32(v2f a, v2f b, v8f c) {
    return __builtin_amdgcn_wmma_f32_16x16x4_f32(
        /*neg_a=*/false, a, /*neg_b=*/false, b,
        /*c_mod=*/(short)0, c, /*reuse_a=*/false, /*reuse_b=*/false);
}

// ---------------- 1) masked mean pooling ----------------
__global__ void __launch_bounds__(256) pool_kernel(const float* __restrict__ x,
                                                   const int* __restrict__ mask,
                                                   float* __restrict__ pooled) {
    const int b = blockIdx.y;
    const int d = blockIdx.x * 256 + threadIdx.x;
    if (d >= DD) return;
    float acc = 0.f, cnt = 0.f;
    const float* xb = x + (size_t)b * SS * DD + d;
    const int* mb = mask + (size_t)b * SS;
    for (int s = 0; s < SS; ++s) {
        float m = (float)mb[s];
        acc += xb[(size_t)s * DD] * m;
        cnt += m;
    }
    pooled[(size_t)b * DD + d] = acc / (cnt + 1e-8f);
}

// ---------------- 2) dense + LayerNorm + leaky_relu(0.01) ----------------
__global__ void __launch_bounds__(256) mlp_ln_kernel(const float* __restrict__ in,
                                                     const float* __restrict__ W,
                                                     const float* __restrict__ bias,
                                                     const float* __restrict__ gamma,
                                                     const float* __restrict__ beta,
                                                     float* __restrict__ out,
                                                     int K, int N) {
    const int b = blockIdx.x;
    const int tid = threadIdx.x;
    __shared__ float ssum[256], ssq[256];
    const float* row = in + (size_t)b * K;
    float yv[4];
    float lsum = 0.f, lsq = 0.f;
    const int per = N >> 8;
    for (int p = 0; p < per; ++p) {
        const int nidx = tid + (p << 8);
        const float* w = W + (size_t)nidx * K;
        float acc = bias[nidx];
        for (int k = 0; k < K; ++k) acc += row[k] * w[k];
        yv[p] = acc;
        lsum += acc; lsq += acc * acc;
    }
    ssum[tid] = lsum; ssq[tid] = lsq;
    __syncthreads();
    for (int off = 128; off > 0; off >>= 1) {
        if (tid < off) { ssum[tid] += ssum[tid + off]; ssq[tid] += ssq[tid + off]; }
        __syncthreads();
    }
    const float mean = ssum[0] / (float)N;
    const float var  = ssq[0] / (float)N - mean * mean;
    const float inv  = rsqrtf(var + 1e-5f);
    for (int p = 0; p < per; ++p) {
        const int nidx = tid + (p << 8);
        float o = (yv[p] - mean) * inv * gamma[nidx] + beta[nidx];
        out[(size_t)b * N + nidx] = (o >= 0.f) ? o : 0.01f * o;
    }
}

// ---------------- 3) mu / logvar / reparameterize ----------------
__global__ void __launch_bounds__(256) latent_kernel(const float* __restrict__ h2,
                                                     const float* __restrict__ W_mu,
                                                     const float* __restrict__ b_mu,
                                                     const float* __restrict__ W_lv,
                                                     const float* __restrict__ b_lv,
                                                     const float* __restrict__ eps,
                                                     float* __restrict__ mu_out,
                                                     float* __restrict__ lv_out,
                                                     float* __restrict__ z) {
    const int b = blockIdx.x, tid = threadIdx.x;
    __shared__ float smu[LL], slv[LL];
    const float* row = h2 + (size_t)b * H2;
    if (tid < LL) {
        const float* w = W_mu + (size_t)tid * H2;
        float acc = b_mu[tid];
        for (int k = 0; k < H2; ++k) acc += row[k] * w[k];
        smu[tid] = acc;
    } else {
        const int l = tid - LL;
        const float* w = W_lv + (size_t)l * H2;
        float acc = b_lv[l];
        for (int k = 0; k < H2; ++k) acc += row[k] * w[k];
        slv[l] = acc;
    }
    __syncthreads();
    if (tid < LL) {
        const float m = smu[tid], lv = slv[tid];
        mu_out[b * LL + tid] = m;
        lv_out[b * LL + tid] = lv;
        z[b * LL + tid] = m + eps[b * LL + tid] * expf(0.5f * lv);
    }
}

// ---------------- 4) h0 = z @ W_l2h^T + b ----------------
__global__ void __launch_bounds__(256) h0_kernel(const float* __restrict__ z,
                                                 const float* __restrict__ W,
                                                 const float* __restrict__ bias,
                                                 float* __restrict__ h0) {
    const int b = blockIdx.x, tid = threadIdx.x;
    const float* row = z + (size_t)b * LL;
    for (int j = tid; j < RH; j += 256) {
        const float* w = W + (size_t)j * LL;
        float acc = bias[j];
        for (int k = 0; k < LL; ++k) acc += row[k] * w[k];
        h0[(size_t)b * RH + j] = acc;
    }
}

// ---------------- 5) x_proj = emb[dec_ids] @ W_ih^T + b_ih (fp32 WMMA) ----
// M = B*T = 32768 (2048 M-tiles), N = 1536 (96 N-tiles), K = 512.
__global__ void __launch_bounds__(256) xproj_wmma_kernel(const int* __restrict__ tgt,
                                                         const float* __restrict__ emb,
                                                         const float* __restrict__ W_ih,
                                                         const float* __restrict__ b_ih,
                                                         float* __restrict__ xp) {
    const int lane = threadIdx.x & 31;
    const int wave = threadIdx.x >> 5;
    const int tile = blockIdx.x * 8 + wave;
    const int ntile = tile % 96;
    const int mtile = tile / 96;
    const int half = lane >> 4, l = lane & 15;

    const int m = mtile * 16 + l;
    const int bb = m >> 9, tt = m & 511;
    const int id = (tt == 0) ? 0 : tgt[bb * TT + tt - 1];  // SOS shift
    const float* arow = emb + (size_t)id * RH;
    const int n = ntile * 16 + l;
    const float* brow = W_ih + (size_t)n * RH;

    v8f acc = {};
    for (int k0 = 0; k0 < RH; k0 += 4) {
        v2f a  = *(const v2f*)(arow + k0 + 2 * half);
        v2f bv = *(const v2f*)(brow + k0 + 2 * half);
        acc = wmma_f32(a, bv, acc);
    }
    const float biasn = b_ih[ntile * 16 + l];
    for (int r = 0; r < 8; ++r) {
        const int mm = mtile * 16 + r + 8 * half;
        xp[(size_t)mm * G3 + ntile * 16 + l] = acc[r] + biasn;
    }
}

// ---------------- 6) persistent GRU scan -------------------------------
// 32 blocks x 384 threads (12 waves). Block i owns h columns [16i, 16i+16).
// W_hh slice (48 rows x 512, ~97 KB) staged once in LDS (bank-conflict-free
// padded stride); per step 12 WMMA tiles -> LDS, fused gates, one atomic
// grid barrier. h double-buffered in global (L2-resident). This step's
// x_proj operands are prefetched (global_prefetch) during the WMMA phase
// so the gate phase hits near-cache instead of eating L2 latency on the
// serial dependency chain.
__global__ void __launch_bounds__(384) gru_kernel(const float* __restrict__ xp,
                                                  const float* __restrict__ W_hh,
                                                  const float* __restrict__ b_hh,
                                                  float* __restrict__ hbuf,   // [2][64][512]
                                                  float* __restrict__ rnn,    // [64][512][512]
                                                  unsigned* __restrict__ sync) {
    extern __shared__ float smem[];
    float* hp   = smem;                 // [3][64][16] = 3072 floats
    float* wlds = smem + 3 * BB * 16;   // 48 rows, stride WSTRIDE

    const int tid  = threadIdx.x;
    const int lane = tid & 31, wave = tid >> 5;   // 12 waves
    const int half = lane >> 4, l = lane & 15;
    const int mtile = wave & 3;       // M tile 0..3
    const int gate  = wave >> 2;      // 0=r, 1=z, 2=n
    const int j0 = blockIdx.x * 16;   // this block's h-column slice

    // stage this block's W_hh rows: gate g, column c -> row g*RH + j0 + c
    for (int i = tid; i < 48 * RH; i += 384) {
        const int rrow = i >> 9;          // i / 512 -> 0..47
        const int kk = i & (RH - 1);
        const int g = rrow >> 4, c = rrow & 15;
        wlds[rrow * WSTRIDE + kk] = W_hh[(size_t)(g * RH + j0 + c) * RH + kk];
    }
    __syncthreads();

    const float* bl = wlds + (gate * 16 + l) * WSTRIDE;   // this lane's B row

    for (int t = 0; t < TT; ++t) {
        const float* hcur = hbuf + (size_t)(t & 1) * (BB * RH);
        float*       hnxt = hbuf + (size_t)((t + 1) & 1) * (BB * RH);

        // prefetch this step's gate operands while the matrix pipe works
        for (int e = tid; e < BB * 16; e += 384) {
            const int br = e >> 4, jj = e & 15;
            const float* xrow = xp + ((size_t)br * TT + t) * G3 + j0 + jj;
            __builtin_prefetch(xrow, 0, 3);
            __builtin_prefetch(xrow + RH, 0, 3);
            __builtin_prefetch(xrow + 2 * RH, 0, 3);
        }

        const float* arow = hcur + (size_t)(mtile * 16 + l) * RH;
        v8f acc = {};
        for (int k0 = 0; k0 < RH; k0 += 4) {
            v2f a  = *(const v2f*)(arow + k0 + 2 * half);   // global (L2)
            v2f bv = *(const v2f*)(bl + k0 + 2 * half);     // LDS
            acc = wmma_f32(a, bv, acc);
        }
        for (int r = 0; r < 8; ++r)
            hp[(gate * BB + mtile * 16 + r + 8 * half) * 16 + l] = acc[r];
        __syncthreads();

        // fused gate computation for 64 x 16 elements
        for (int e = tid; e < BB * 16; e += 384) {
            const int br = e >> 4, jj = e & 15;
            const int j  = j0 + jj;
            const float* xrow = xp + ((size_t)br * TT + t) * G3;
            const float hr = hp[(0 * BB + br) * 16 + jj] + b_hh[j];
            const float hz = hp[(1 * BB + br) * 16 + jj] + b_hh[RH + j];
            const float hn = hp[(2 * BB + br) * 16 + jj] + b_hh[2 * RH + j];
            const float rg = 1.f / (1.f + expf(-(xrow[j] + hr)));
            const float zg = 1.f / (1.f + expf(-(xrow[RH + j] + hz)));
            const float ng = tanhf(xrow[2 * RH + j] + rg * hn);
            const float hold = hcur[(size_t)br * RH + j];
            const float hnew = (1.f - zg) * ng + zg * hold;
            hnxt[(size_t)br * RH + j] = hnew;
            rnn[((size_t)br * TT + t) * RH + j] = hnew;
        }
        __syncthreads();

        // grid barrier (generation counter)
        if (tid == 0) {
            __threadfence();
            const unsigned old = atomicAdd(&sync[0], 1u);
            if (old == GRU_BLOCKS - 1) {
                atomicExch(&sync[0], 0u);
                __threadfence();
                atomicAdd(&sync[1], 1u);
            } else {
                while (atomicAdd(&sync[1], 0u) < (unsigned)(t + 1)) {
                    __builtin_amdgcn_s_sleep(8);
                }
            }
            __threadfence();
        }
        __syncthreads();
    }
}

// ---------------- 7) logits = rnn_out @ W_out^T + b_out (fp32 WMMA) -----
// M = 32768 (2048 M-tiles), N = 33 padded to 3 N-tiles, K = 512.
__global__ void __launch_bounds__(256) logits_wmma_kernel(const float* __restrict__ rnn,
                                                          const float* __restrict__ W_out,
                                                          const float* __restrict__ b_out,
                                                          float* __restrict__ out) {
    const int lane = threadIdx.x & 31;
    const int wave = threadIdx.x >> 5;
    const int tile = blockIdx.x * 8 + wave;     // < 2048*3
    const int ntile = tile % 3;
    const int mtile = tile / 3;
    const int half = lane >> 4, l = lane & 15;

    const float* arow = rnn + (size_t)(mtile * 16 + l) * RH;
    const int nn = ntile * 16 + l;
    const int nload = (nn < VV) ? nn : (VV - 1);   // clamp B row for padding
    const float* brow = W_out + (size_t)nload * RH;

    v8f acc = {};
    for (int k0 = 0; k0 < RH; k0 += 4) {
        v2f a  = *(const v2f*)(arow + k0 + 2 * half);
        v2f bv = *(const v2f*)(brow + k0 + 2 * half);
        acc = wmma_f32(a, bv, acc);
    }
    if (nn < VV) {
        const float biasn = b_out[nn];
        for (int r = 0; r < 8; ++r) {
            const int mm = mtile * 16 + r + 8 * half;
            out[(size_t)mm * VV + nn] = acc[r] + biasn;
        }
    }
}

extern "C" void kernel_launch(void* const* d_in, const int* in_sizes, int n_in,
                              void* d_out, int out_size, void* d_ws, size_t ws_size,
                              hipStream_t stream) {
    const float* x        = (const float*)d_in[0];
    const int*   amask    = (const int*)d_in[1];
    const int*   tgt      = (const int*)d_in[2];
    const float* eps      = (const float*)d_in[3];
    const float* enc_W1   = (const float*)d_in[4];
    const float* enc_b1   = (const float*)d_in[5];
    const float* enc_g1   = (const float*)d_in[6];
    const float* enc_be1  = (const float*)d_in[7];
    const float* enc_W2   = (const float*)d_in[8];
    const float* enc_b2   = (const float*)d_in[9];
    const float* enc_g2   = (const float*)d_in[10];
    const float* enc_be2  = (const float*)d_in[11];
    const float* W_mu     = (const float*)d_in[12];
    const float* b_mu     = (const float*)d_in[13];
    const float* W_lv     = (const float*)d_in[14];
    const float* b_lv     = (const float*)d_in[15];
    const float* W_l2h    = (const float*)d_in[16];
    const float* b_l2h    = (const float*)d_in[17];
    const float* emb      = (const float*)d_in[18];
    const float* W_ih     = (const float*)d_in[19];
    const float* b_ih     = (const float*)d_in[20];
    const float* W_hh     = (const float*)d_in[21];
    const float* b_hh     = (const float*)d_in[22];
    const float* W_out    = (const float*)d_in[23];
    const float* b_out    = (const float*)d_in[24];

    float* ws = (float*)d_ws;
    size_t off = 0;
    float* pooled = ws + off; off += (size_t)BB * DD;
    float* h1     = ws + off; off += (size_t)BB * H1;
    float* h2     = ws + off; off += (size_t)BB * H2;
    float* z      = ws + off; off += (size_t)BB * LL;
    float* hbuf   = ws + off; off += (size_t)2 * BB * RH;
    float* xpb    = ws + off; off += (size_t)BB * TT * G3;
    float* rnn    = ws + off; off += (size_t)BB * TT * RH;
    unsigned* syn = (unsigned*)(ws + off);

    float* out_logits = (float*)d_out;                       // [B,T,V]
    float* out_mu     = out_logits + (size_t)BB * TT * VV;   // [B,L]
    float* out_lv     = out_mu + (size_t)BB * LL;            // [B,L]

    pool_kernel<<<dim3((DD + 255) / 256, BB), 256, 0, stream>>>(x, amask, pooled);
    mlp_ln_kernel<<<BB, 256, 0, stream>>>(pooled, enc_W1, enc_b1, enc_g1, enc_be1, h1, DD, H1);
    mlp_ln_kernel<<<BB, 256, 0, stream>>>(h1, enc_W2, enc_b2, enc_g2, enc_be2, h2, H1, H2);
    latent_kernel<<<BB, 256, 0, stream>>>(h2, W_mu, b_mu, W_lv, b_lv, eps, out_mu, out_lv, z);
    h0_kernel<<<BB, 256, 0, stream>>>(z, W_l2h, b_l2h, hbuf);
    xproj_wmma_kernel<<<(2048 * 96) / 8, 256, 0, stream>>>(tgt, emb, W_ih, b_ih, xpb);
    hipMemsetAsync(syn, 0, 2 * sizeof(unsigned), stream);
    const size_t gru_lds = (size_t)(3 * BB * 16 + 48 * WSTRIDE) * sizeof(float); // ~111 KB
    gru_kernel<<<GRU_BLOCKS, 384, gru_lds, stream>>>(xpb, W_hh, b_hh, hbuf, rnn, syn);
    logits_wmma_kernel<<<(2048 * 3) / 8, 256, 0, stream>>>(rnn, W_out, b_out, out_logits);
}